// GNN_node_56435870269808
// MI455X (gfx1250) — compile-verified
//
#include <hip/hip_runtime.h>
#include <hip/hip_bf16.h>

typedef float v2f __attribute__((ext_vector_type(2)));
typedef float v8f __attribute__((ext_vector_type(8)));

#define D 128
#define NUM_LAYERS 4

// ---------------------------------------------------------------------------
// AtomEncoder: h[n,d] = sum_{c<9} atom_emb[c, x_feat[n,c], d]
// ---------------------------------------------------------------------------
__global__ __launch_bounds__(256) void atom_encoder_kernel(
    const int* __restrict__ x_feat,      // [N,9]
    const float* __restrict__ atom_emb,  // [9,128,D]
    float* __restrict__ h, int N) {
  int t = blockIdx.x * blockDim.x + threadIdx.x;
  if (t >= N * D) return;
  int n = t >> 7, d = t & (D - 1);
  float acc = 0.f;
#pragma unroll
  for (int c = 0; c < 9; ++c) {
    int idx = x_feat[n * 9 + c];
    acc += atom_emb[((size_t)(c * 128 + idx)) * D + d];
  }
  h[t] = acc;
}

// ---------------------------------------------------------------------------
// Edge message + scatter-add: agg[dst] += relu(h[src] + bond_emb(ea))
// 32 lanes per edge, float4 per lane. agg (51MB) lives in L2 -> fast atomics.
// ---------------------------------------------------------------------------
__global__ __launch_bounds__(256) void edge_msg_kernel(
    const float* __restrict__ h,
    const int* __restrict__ ei,    // [2,E]
    const int* __restrict__ ea,    // [E,3]
    const float* __restrict__ bond, // [3,8,D] for this layer
    float* __restrict__ agg, long long E) {
  long long t = (long long)blockIdx.x * blockDim.x + threadIdx.x;
  if (t >= E * 32) return;
  int e = (int)(t >> 5);
  int j = (int)(t & 31);
  int d0 = j << 2;
  int src = ei[e];
  int dst = ei[E + e];
  const float4 hv = *reinterpret_cast<const float4*>(h + (size_t)src * D + d0);
  float sx = hv.x, sy = hv.y, sz = hv.z, sw = hv.w;
#pragma unroll
  for (int c = 0; c < 3; ++c) {
    int idx = ea[e * 3 + c];
    const float4 be =
        *reinterpret_cast<const float4*>(bond + ((size_t)(c * 8 + idx)) * D + d0);
    sx += be.x; sy += be.y; sz += be.z; sw += be.w;
  }
  sx = fmaxf(sx, 0.f); sy = fmaxf(sy, 0.f);
  sz = fmaxf(sz, 0.f); sw = fmaxf(sw, 0.f);
  float* a = agg + (size_t)dst * D + d0;
  atomicAdd(a + 0, sx);
  atomicAdd(a + 1, sy);
  atomicAdd(a + 2, sz);
  atomicAdd(a + 3, sw);
}

// ---------------------------------------------------------------------------
// GEMM1 (WMMA f32 16x16x4): y1[N,256] = ((1+eps)*h + agg) @ W1 + b1
// Block = 128 threads = 4 waves; block owns 16 rows, wave owns a 16-col tile.
// A tile staged in LDS (padded stride), fused z computation on load.
// ---------------------------------------------------------------------------
__global__ __launch_bounds__(128) void gemm1_wmma_kernel(
    const float* __restrict__ h, const float* __restrict__ agg,
    const float* __restrict__ epsp,
    const float* __restrict__ W1,  // [128,256] row-major
    const float* __restrict__ b1,  // [256]
    float* __restrict__ y1, int N) {
  __shared__ float As[16][132];  // pad 128->132 to avoid bank conflicts
  const int tid = threadIdx.x;
  const int m0 = blockIdx.x * 16;
  const float ep = 1.0f + epsp[0];
  for (int i = tid; i < 16 * 128; i += 128) {
    int r = i >> 7, k = i & 127;
    size_t g = (size_t)(m0 + r) * D + k;
    As[r][k] = ep * h[g] + agg[g];
  }
  __syncthreads();
  const int wave = tid >> 5, lane = tid & 31;
  const int hi = lane >> 4, lr = lane & 15;
  const int n0 = (blockIdx.y * 4 + wave) * 16;
  v8f acc = {0.f, 0.f, 0.f, 0.f, 0.f, 0.f, 0.f, 0.f};
  for (int k = 0; k < 128; k += 4) {
    const int ka = k + hi * 2;  // lanes0-15: K=k,k+1 ; lanes16-31: K=k+2,k+3
    v2f a, b;
    a.x = As[lr][ka];
    a.y = As[lr][ka + 1];
    b.x = W1[(size_t)ka * 256 + n0 + lr];
    b.y = W1[(size_t)(ka + 1) * 256 + n0 + lr];
    acc = __builtin_amdgcn_wmma_f32_16x16x4_f32(false, a, false, b, (short)0,
                                                acc, false, false);
  }
  const float bias = b1[n0 + lr];
#pragma unroll
  for (int v = 0; v < 8; ++v) {
    int row = m0 + v + hi * 8;  // C/D layout: lanes16-31 hold M=v+8
    y1[(size_t)row * 256 + n0 + lr] = acc[v] + bias;
  }
}

// ---------------------------------------------------------------------------
// GEMM2 (WMMA f32 16x16x4): y2[N,128] = relu(y1*scale+shift) @ W2 + b2
// BN1 apply + ReLU fused into A-tile load.
// ---------------------------------------------------------------------------
__global__ __launch_bounds__(128) void gemm2_wmma_kernel(
    const float* __restrict__ y1, const float* __restrict__ scale,
    const float* __restrict__ shift,
    const float* __restrict__ W2,  // [256,128] row-major
    const float* __restrict__ b2,  // [128]
    float* __restrict__ y2, int N) {
  __shared__ float As[16][260];  // 16 x 256 padded
  const int tid = threadIdx.x;
  const int m0 = blockIdx.x * 16;
  for (int i = tid; i < 16 * 256; i += 128) {
    int r = i >> 8, k = i & 255;
    float v = y1[(size_t)(m0 + r) * 256 + k] * scale[k] + shift[k];
    As[r][k] = fmaxf(v, 0.f);
  }
  __syncthreads();
  const int wave = tid >> 5, lane = tid & 31;
  const int hi = lane >> 4, lr = lane & 15;
  const int n0 = (blockIdx.y * 4 + wave) * 16;
  v8f acc = {0.f, 0.f, 0.f, 0.f, 0.f, 0.f, 0.f, 0.f};
  for (int k = 0; k < 256; k += 4) {
    const int ka = k + hi * 2;
    v2f a, b;
    a.x = As[lr][ka];
    a.y = As[lr][ka + 1];
    b.x = W2[(size_t)ka * 128 + n0 + lr];
    b.y = W2[(size_t)(ka + 1) * 128 + n0 + lr];
    acc = __builtin_amdgcn_wmma_f32_16x16x4_f32(false, a, false, b, (short)0,
                                                acc, false, false);
  }
  const float bias = b2[n0 + lr];
#pragma unroll
  for (int v = 0; v < 8; ++v) {
    int row = m0 + v + hi * 8;
    y2[(size_t)row * 128 + n0 + lr] = acc[v] + bias;
  }
}

// ---------------------------------------------------------------------------
// BN column statistics: coalesced per-column partial sums + global atomics
// blockDim.x == C (128 or 256); each block covers 64 rows.
// ---------------------------------------------------------------------------
__global__ void bnstats_kernel(const float* __restrict__ y, int N, int C,
                               float* __restrict__ sum,
                               float* __restrict__ sumsq) {
  int col = threadIdx.x;
  int row0 = blockIdx.x * 64;
  int rmax = N - row0; if (rmax > 64) rmax = 64;
  float s = 0.f, s2 = 0.f;
  for (int r = 0; r < rmax; ++r) {
    float v = y[(size_t)(row0 + r) * C + col];
    s += v;
    s2 += v * v;
  }
  atomicAdd(&sum[col], s);
  atomicAdd(&sumsq[col], s2);
}

__global__ void bnfinal_kernel(const float* __restrict__ sum,
                               const float* __restrict__ sumsq,
                               const float* __restrict__ gamma,
                               const float* __restrict__ beta,
                               float* __restrict__ scale,
                               float* __restrict__ shift, int N, int C) {
  int c = threadIdx.x;
  if (c >= C) return;
  float invN = 1.0f / (float)N;
  float mean = sum[c] * invN;
  float var = sumsq[c] * invN - mean * mean;
  float inv = rsqrtf(var + 1e-5f);
  float sc = gamma[c] * inv;
  scale[c] = sc;
  shift[c] = beta[c] - sc * mean;
}

// out = maybe_relu(y * scale + shift), C == 128
__global__ __launch_bounds__(256) void bnapply_kernel(
    const float* __restrict__ y, const float* __restrict__ scale,
    const float* __restrict__ shift, float* __restrict__ out, int total,
    int do_relu) {
  int t = blockIdx.x * blockDim.x + threadIdx.x;
  if (t >= total) return;
  int c = t & (D - 1);
  float v = y[t] * scale[c] + shift[c];
  if (do_relu) v = fmaxf(v, 0.f);
  out[t] = v;
}

// ---------------------------------------------------------------------------
extern "C" void kernel_launch(void* const* d_in, const int* in_sizes, int n_in,
                              void* d_out, int out_size, void* d_ws,
                              size_t ws_size, hipStream_t stream) {
  const int* x_feat      = (const int*)d_in[0];
  const int* edge_index  = (const int*)d_in[1];
  const int* edge_attr   = (const int*)d_in[2];
  const int* cayley_ei   = (const int*)d_in[3];
  const int* cayley_ea   = (const int*)d_in[4];
  const float* atom_emb  = (const float*)d_in[5];
  const float* bond_emb  = (const float*)d_in[6];
  const float* eps       = (const float*)d_in[7];
  const float* W1        = (const float*)d_in[8];
  const float* b1        = (const float*)d_in[9];
  const float* bn1_gamma = (const float*)d_in[10];
  const float* bn1_beta  = (const float*)d_in[11];
  const float* W2        = (const float*)d_in[12];
  const float* b2        = (const float*)d_in[13];
  const float* bn2_gamma = (const float*)d_in[14];
  const float* bn2_beta  = (const float*)d_in[15];

  const int N = in_sizes[0] / 9;          // 100000 (divisible by 16)
  const long long E = in_sizes[1] / 2;    // 600000
  const long long EC = in_sizes[3] / 2;   // 500000
  const int total = N * D;

  // Workspace: h | agg(=y2) | y1 | stats  -> ~205 MB + 4KB
  float* h     = (float*)d_ws;
  float* agg   = h + (size_t)N * D;
  float* y1    = agg + (size_t)N * D;
  float* stats = y1 + (size_t)N * 256;
  float* sum   = stats;
  float* sumsq = stats + 256;
  float* scale = stats + 512;
  float* shift = stats + 768;

  atom_encoder_kernel<<<(total + 255) / 256, 256, 0, stream>>>(x_feat, atom_emb,
                                                               h, N);

  const int statBlocks = (N + 63) / 64;
  for (int layer = 0; layer < NUM_LAYERS; ++layer) {
    const int* ei; const int* ea; long long ne;
    if (layer & 1) { ei = cayley_ei; ea = cayley_ea; ne = EC; }
    else           { ei = edge_index; ea = edge_attr; ne = E; }

    // scatter-add aggregation
    hipMemsetAsync(agg, 0, (size_t)N * D * sizeof(float), stream);
    long long tE = ne * 32;
    edge_msg_kernel<<<(unsigned)((tE + 255) / 256), 256, 0, stream>>>(
        h, ei, ea, bond_emb + (size_t)layer * 3 * 8 * D, agg, ne);

    // GEMM1: [N,128] x [128,256]
    dim3 g1(N / 16, 4);
    gemm1_wmma_kernel<<<g1, 128, 0, stream>>>(
        h, agg, eps + layer, W1 + (size_t)layer * 128 * 256,
        b1 + (size_t)layer * 256, y1, N);

    // BN1 stats -> scale/shift
    hipMemsetAsync(sum, 0, 512 * sizeof(float), stream);
    bnstats_kernel<<<statBlocks, 256, 0, stream>>>(y1, N, 256, sum, sumsq);
    bnfinal_kernel<<<1, 256, 0, stream>>>(sum, sumsq, bn1_gamma + layer * 256,
                                          bn1_beta + layer * 256, scale, shift,
                                          N, 256);

    // GEMM2: relu(BN1(y1)) x [256,128]  (BN apply fused into A-load)
    float* y2 = agg;  // agg no longer needed this layer
    dim3 g2(N / 16, 2);
    gemm2_wmma_kernel<<<g2, 128, 0, stream>>>(
        y1, scale, shift, W2 + (size_t)layer * 256 * 128,
        b2 + (size_t)layer * 128, y2, N);

    // BN2 stats -> apply (+ReLU except last layer)
    hipMemsetAsync(sum, 0, 512 * sizeof(float), stream);
    bnstats_kernel<<<statBlocks, 128, 0, stream>>>(y2, N, 128, sum, sumsq);
    bnfinal_kernel<<<1, 128, 0, stream>>>(sum, sumsq, bn2_gamma + layer * 128,
                                          bn2_beta + layer * 128, scale, shift,
                                          N, 128);
    float* dst = (layer == NUM_LAYERS - 1) ? (float*)d_out : h;
    bnapply_kernel<<<(total + 255) / 256, 256, 0, stream>>>(
        y2, scale, shift, dst, total, (layer < NUM_LAYERS - 1) ? 1 : 0);
  }
}